// PairToPair_4063039062768
// MI455X (gfx1250) — compile-verified
//
#include <hip/hip_runtime.h>
#include <hip/hip_bf16.h>
#include <math.h>

#define NN 1024
#define KK 32
#define AA 5
#define PP 128
#define FIN 158   // P + 2*A*3   (padded to 160)
#define FPP 278   // P + 2*A*A*3 (padded to 280)

typedef float v2f __attribute__((ext_vector_type(2)));
typedef float v8f __attribute__((ext_vector_type(8)));

__device__ __forceinline__ v8f wmma4(v2f a, v2f b, v8f c) {
  // D = A(16x4 f32) * B(4x16 f32) + C(16x16 f32), full fp32 precision
  return __builtin_amdgcn_wmma_f32_16x16x4_f32(false, a, false, b, (short)0, c, false, false);
}

__device__ __forceinline__ float softplus_f(float x) {
  return fmaxf(x, 0.0f) + log1pf(expf(-fabsf(x)));
}

__device__ __forceinline__ float gelu_tanh(float x) {
  const float c = 0.7978845608028654f; // sqrt(2/pi)
  return 0.5f * x * (1.0f + tanhf(c * (x + 0.044715f * x * x * x)));
}

// ---------------------------------------------------------------------------
// Kernel 0: pack weights for WMMA B-fragments.
// Wp[r2*128+col] = {W[2*r2][col], W[2*r2+1][col]}, zero-padded past `rows`.
// One aligned b64 load then feeds a 4x16 B fragment half directly.
// ---------------------------------------------------------------------------
__global__ void k_pack(const float* __restrict__ W, int rows,
                       v2f* __restrict__ Wp, int r2max) {
  const int idx = blockIdx.x * 256 + threadIdx.x;  // r2*128 + col
  if (idx >= r2max * PP) return;
  const int r2 = idx / PP, col = idx % PP;
  v2f v;
  v.x = (2 * r2     < rows) ? W[(2 * r2)     * PP + col] : 0.0f;
  v.y = (2 * r2 + 1 < rows) ? W[(2 * r2 + 1) * PP + col] : 0.0f;
  Wp[idx] = v;
}

// ---------------------------------------------------------------------------
// Kernel 1: Gram-Schmidt frames + local neighbour coordinates
// local[n,k,a,:] = R[n] @ (pos[nbr[n,k],a,:] - CA[n])
// grid = N blocks, block = 160 threads (K*A)
// ---------------------------------------------------------------------------
__global__ void k_frames_local(const float* __restrict__ pos,
                               const int* __restrict__ nbr,
                               float* __restrict__ localv) {
  __shared__ float R[9];
  __shared__ float CAs[3];
  const int n = blockIdx.x;
  if (threadIdx.x == 0) {
    const float* pn = pos + (size_t)n * AA * 3;
    float nax = pn[0], nay = pn[1], naz = pn[2];
    float cax = pn[3], cay = pn[4], caz = pn[5];
    float cx  = pn[6], cy  = pn[7], cz  = pn[8];
    float e1x = cx - cax, e1y = cy - cay, e1z = cz - caz;
    float n1 = sqrtf(e1x*e1x + e1y*e1y + e1z*e1z) + 1e-6f;
    e1x /= n1; e1y /= n1; e1z /= n1;
    float v2x = nax - cax, v2y = nay - cay, v2z = naz - caz;
    float d = v2x*e1x + v2y*e1y + v2z*e1z;
    float e2x = v2x - d*e1x, e2y = v2y - d*e1y, e2z = v2z - d*e1z;
    float n2 = sqrtf(e2x*e2x + e2y*e2y + e2z*e2z) + 1e-6f;
    e2x /= n2; e2y /= n2; e2z /= n2;
    float e3x = e1y*e2z - e1z*e2y;
    float e3y = e1z*e2x - e1x*e2z;
    float e3z = e1x*e2y - e1y*e2x;
    R[0]=e1x; R[1]=e1y; R[2]=e1z;
    R[3]=e2x; R[4]=e2y; R[5]=e2z;
    R[6]=e3x; R[7]=e3y; R[8]=e3z;
    CAs[0]=cax; CAs[1]=cay; CAs[2]=caz;
  }
  __syncthreads();
  const int t = threadIdx.x;
  if (t < KK * AA) {
    const int k = t / AA, a = t % AA;
    const int nb = nbr[n * KK + k];
    const float* p = pos + ((size_t)nb * AA + a) * 3;
    float dx = p[0] - CAs[0], dy = p[1] - CAs[1], dz = p[2] - CAs[2];
    float* o = localv + (((size_t)n * KK + k) * AA + a) * 3;
    o[0] = R[0]*dx + R[1]*dy + R[2]*dz;
    o[1] = R[3]*dx + R[4]*dy + R[5]*dz;
    o[2] = R[6]*dx + R[7]*dy + R[8]*dz;
  }
}

// ---------------------------------------------------------------------------
// Kernel 2: left/right projections.
// rows = (n,k) flat (32768), feats = [pair(128) | dscale*local(15) | dir(15)] pad->160
// Each block: 16 rows; 8 waves, each owns a 16-col slice; dual WMMA accumulators.
// grid = 2048 blocks, block = 256 threads
// ---------------------------------------------------------------------------
__global__ void __launch_bounds__(256)
k_leftright(const float* __restrict__ pair,
            const float* __restrict__ localv,
            const v2f* __restrict__ Wlp,
            const v2f* __restrict__ Wrp,
            const float* __restrict__ dscp,
            float* __restrict__ leftB,
            float* __restrict__ rightB) {
  __shared__ float F[16][160];
  const int b = blockIdx.x;
  const int n = (b * 16) / KK;        // 16-row tiles align within one n (K=32)
  const int kbase = (b * 16) % KK;    // 0 or 16
  const float dscale = softplus_f(dscp[0]) * 0.1f;

  for (int e = threadIdx.x; e < 16 * 160; e += 256) {
    const int r = e / 160, col = e % 160;
    const int k = kbase + r;
    float val;
    if (col < PP) {
      val = pair[(((size_t)n * KK + k) * PP) + col];
    } else if (col < PP + 15) {
      const int f = col - PP;
      val = dscale * localv[(((size_t)n * KK + k) * AA) * 3 + f];
    } else if (col < PP + 30) {
      const int f = col - (PP + 15);
      const int a = f / 3, c = f % 3;
      const float* L = localv + (((size_t)n * KK + k) * AA + a) * 3;
      float nr = sqrtf(L[0]*L[0] + L[1]*L[1] + L[2]*L[2]) + 1e-6f;
      val = L[c] / nr;
    } else {
      val = 0.0f;  // zero-pad 158 -> 160
    }
    F[r][col] = val;
  }
  __syncthreads();

  const int lane = threadIdx.x & 31;
  const int w = threadIdx.x >> 5;
  const int half = lane >> 4;
  const int lcol = lane & 15;
  const int colg = w * 16 + lcol;

  v8f cl = {}; v8f cr = {};
  #pragma unroll 4
  for (int kq = 0; kq < 40; ++kq) {                      // K = 160, no tail
    const v2f a  = *(const v2f*)&F[lcol][kq * 4 + 2 * half];
    const v2f bl = Wlp[(2 * kq + half) * PP + colg];     // single b64 load
    const v2f br = Wrp[(2 * kq + half) * PP + colg];
    cl = wmma4(a, bl, cl);
    cr = wmma4(a, br, cr);
  }

  // C/D layout: vgpr i -> row i (lanes 0-15) / row i+8 (lanes 16-31)
  #pragma unroll
  for (int i = 0; i < 8; ++i) {
    const int r = i + 8 * half;
    const int k = kbase + r;
    leftB[(((size_t)n * KK + k) * PP) + colg] = cl[i];
    if (k < 8) rightB[(((size_t)n * 8 + k) * PP) + colg] = cr[i];
  }
}

// ---------------------------------------------------------------------------
// Kernel 3: fused pair-pair block.
// One block per 16-row tile of (k,j): tile t covers k in {2t, 2t+1}, j = 0..7.
// grid = N*16 = 16384 blocks, block = 256 threads (8 waves).
// ---------------------------------------------------------------------------
__global__ void __launch_bounds__(256)
k_pairpair(const float* __restrict__ localv,
           const float* __restrict__ leftB,
           const float* __restrict__ rightB,
           const v2f* __restrict__ Wgp,
           const v2f* __restrict__ Whp,
           const v2f* __restrict__ Wop,
           const float* __restrict__ pm,
           const float* __restrict__ dscp,
           float* __restrict__ out) {
  __shared__ float PPF[16][280];
  __shared__ float HID[16][PP];
  const int b = blockIdx.x;
  const int n = b >> 4;
  const int t = b & 15;
  const int k0 = t * 2;
  const float dscale = softplus_f(dscp[0]) * 0.1f;

  // Build ppf tile: row r -> (k = k0 + (r>>3), j = r&7)
  for (int e = threadIdx.x; e < 16 * 280; e += 256) {
    const int r = e / 280, col = e % 280;
    const int k = k0 + (r >> 3);
    const int j = r & 7;
    float val;
    if (col < PP) {
      val = leftB[(((size_t)n * KK + k) * PP) + col] +
            rightB[(((size_t)n * 8 + j) * PP) + col];
    } else if (col < FPP) {
      int f = col - PP;
      int isdir = 0;
      if (f >= 75) { f -= 75; isdir = 1; }
      const int a = f / 15, rem = f % 15, bb = rem / 3, c = rem % 3;
      const float* LK = localv + (((size_t)n * KK + k) * AA + a) * 3;
      const float* LJ = localv + (((size_t)n * KK + j) * AA + bb) * 3;
      const float rc = LK[c] - LJ[c];
      if (!isdir) {
        val = dscale * rc;
      } else {
        float rx = LK[0]-LJ[0], ry = LK[1]-LJ[1], rz = LK[2]-LJ[2];
        float nr = sqrtf(rx*rx + ry*ry + rz*rz) + 1e-6f;
        val = rc / nr;
      }
    } else {
      val = 0.0f;  // zero-pad 278 -> 280
    }
    PPF[r][col] = val;
  }
  __syncthreads();

  const int lane = threadIdx.x & 31;
  const int w = threadIdx.x >> 5;
  const int half = lane >> 4;
  const int lcol = lane & 15;
  const int colg = w * 16 + lcol;

  // GEMMs: gate = ppf@Wg, hid0 = ppf@Wh  (K = 280, 70 steps of 4, no tail)
  v8f cg = {}; v8f ch = {};
  #pragma unroll 5
  for (int kq = 0; kq < 70; ++kq) {
    const v2f a  = *(const v2f*)&PPF[lcol][kq * 4 + 2 * half];
    const v2f bg = Wgp[(2 * kq + half) * PP + colg];
    const v2f bh = Whp[(2 * kq + half) * PP + colg];
    cg = wmma4(a, bg, cg);
    ch = wmma4(a, bh, ch);
  }

  // hidden = gelu(gate) * hid0 -> LDS
  #pragma unroll
  for (int i = 0; i < 8; ++i) {
    HID[i + 8 * half][colg] = gelu_tanh(cg[i]) * ch[i];
  }
  __syncthreads();

  // out = hidden @ Wo  (K = 128, 32 steps of 4)
  v8f co = {};
  #pragma unroll 4
  for (int kq = 0; kq < 32; ++kq) {
    const v2f a  = *(const v2f*)&HID[lcol][kq * 4 + 2 * half];
    const v2f bo = Wop[(2 * kq + half) * PP + colg];
    co = wmma4(a, bo, co);
  }

  // Masked mean over j. Accumulator rows: lanes 0-15 hold (k=k0, j=i),
  // lanes 16-31 hold (k=k0+1, j=i) -> the j-reduction is in-register.
  const int k = k0 + half;
  const float pmk = pm[n * KK + k];
  float s = 0.0f, cnt = 0.0f;
  #pragma unroll
  for (int i = 0; i < 8; ++i) {
    const float m = pmk * pm[n * KK + i];
    if (m > 0.0f) { s += co[i]; cnt += m; }
  }
  out[(((size_t)n * KK + k) * PP) + colg] = s / fmaxf(cnt, 1.0f);
}

// ---------------------------------------------------------------------------
extern "C" void kernel_launch(void* const* d_in, const int* in_sizes, int n_in,
                              void* d_out, int out_size, void* d_ws, size_t ws_size,
                              hipStream_t stream) {
  const float* pair  = (const float*)d_in[0];
  const float* pos   = (const float*)d_in[1];
  const int*   nbr   = (const int*)d_in[2];
  const float* pmask = (const float*)d_in[3];
  const float* dscp  = (const float*)d_in[4];
  const float* Wl = (const float*)d_in[5];
  const float* Wr = (const float*)d_in[6];
  const float* Wg = (const float*)d_in[7];
  const float* Wh = (const float*)d_in[8];
  const float* Wo = (const float*)d_in[9];
  float* outp = (float*)d_out;

  // Workspace layout (floats / v2f, all 8B-aligned)
  float* ws = (float*)d_ws;
  float* localv = ws;                                   // N*K*A*3   = 491520 f
  float* leftB  = localv + (size_t)NN * KK * AA * 3;    // N*K*P     = 4194304 f
  float* rightB = leftB + (size_t)NN * KK * PP;         // N*8*P     = 1048576 f
  v2f* Wlp = (v2f*)(rightB + (size_t)NN * 8 * PP);      // 80*128  v2f
  v2f* Wrp = Wlp + 80 * PP;                             // 80*128  v2f
  v2f* Wgp = Wrp + 80 * PP;                             // 140*128 v2f
  v2f* Whp = Wgp + 140 * PP;                            // 140*128 v2f
  v2f* Wop = Whp + 140 * PP;                            // 64*128  v2f

  // Pack weights (zero-padded) into B-fragment-friendly row-pair layout.
  k_pack<<<(80  * PP + 255) / 256, 256, 0, stream>>>(Wl, FIN, Wlp, 80);
  k_pack<<<(80  * PP + 255) / 256, 256, 0, stream>>>(Wr, FIN, Wrp, 80);
  k_pack<<<(140 * PP + 255) / 256, 256, 0, stream>>>(Wg, FPP, Wgp, 140);
  k_pack<<<(140 * PP + 255) / 256, 256, 0, stream>>>(Wh, FPP, Whp, 140);
  k_pack<<<(64  * PP + 255) / 256, 256, 0, stream>>>(Wo, PP,  Wop, 64);

  k_frames_local<<<NN, KK * AA, 0, stream>>>(pos, nbr, localv);
  k_leftright<<<(NN * KK) / 16, 256, 0, stream>>>(pair, localv, Wlp, Wrp, dscp,
                                                  leftB, rightB);
  k_pairpair<<<NN * 16, 256, 0, stream>>>(localv, leftB, rightB, Wgp, Whp, Wop,
                                          pmask, dscp, outp);
}